// EmotionClassifier_60009283059738
// MI455X (gfx1250) — compile-verified
//
#include <hip/hip_runtime.h>
#include <hip/hip_bf16.h>

// Shapes from the reference
#define LL 13
#define BB 16
#define SS 1000
#define DD 1024
#define HH 256
#define NLAB 8

#define CHUNK 50
#define NCHUNK 20   // CHUNK * NCHUNK == SS

// d_out layout (floats): pred[BB*NLAB] | final_emb[BB*DD] | dyn_w[BB*LL]
#define OUT_PRED 0
#define OUT_FE   (BB * NLAB)
#define OUT_DW   (BB * NLAB + BB * DD)

typedef __attribute__((ext_vector_type(2))) float v2f;
typedef __attribute__((ext_vector_type(4))) float v4f;
typedef __attribute__((ext_vector_type(8))) float v8f;

// ---------------------------------------------------------------------------
// Stage 1: partial masked sums over S-chunks.
// grid: (L*B, NCHUNK), block: 256 threads, each thread owns 4 consecutive d.
// Streams the 852 MB `feature` tensor with coalesced non-temporal float4 loads.
// ---------------------------------------------------------------------------
__global__ __launch_bounds__(256) void pool_partial_kernel(
    const float* __restrict__ feature, const int* __restrict__ lens,
    float* __restrict__ partials) {
  const int lb = blockIdx.x;          // l * BB + b
  const int c  = blockIdx.y;          // s-chunk
  const int l  = lb / BB;
  const int b  = lb % BB;
  const int t  = threadIdx.x;
  const int d0 = t * 4;

  const int len = lens[b];
  const int s0  = c * CHUNK;
  const int s1  = (s0 + CHUNK < len) ? (s0 + CHUNK) : len;

  v4f acc = {0.f, 0.f, 0.f, 0.f};
  const size_t base = ((size_t)l * BB + b) * (size_t)SS * DD + (size_t)d0;
  for (int s = s0; s < s1; ++s) {
    const v4f* p = (const v4f*)(feature + base + (size_t)s * DD);
    v4f v = __builtin_nontemporal_load(p);
    acc.x += v.x; acc.y += v.y; acc.z += v.z; acc.w += v.w;
  }
  float* dst = partials + ((size_t)lb * NCHUNK + c) * DD + d0;
  dst[0] = acc.x; dst[1] = acc.y; dst[2] = acc.z; dst[3] = acc.w;
}

// ---------------------------------------------------------------------------
// Stage 2: reduce NCHUNK partials, divide by len -> pooled stored b-major:
// pooled[(b*LL + l) * DD + d]
// ---------------------------------------------------------------------------
__global__ __launch_bounds__(256) void pool_reduce_kernel(
    const float* __restrict__ partials, const int* __restrict__ lens,
    float* __restrict__ pooled) {
  const int lb = blockIdx.x;          // l * BB + b
  const int l  = lb / BB;
  const int b  = lb % BB;
  const int t  = threadIdx.x;
  const int d0 = t * 4;

  v4f acc = {0.f, 0.f, 0.f, 0.f};
  for (int c = 0; c < NCHUNK; ++c) {
    const float* p = partials + ((size_t)lb * NCHUNK + c) * DD + d0;
    acc.x += p[0]; acc.y += p[1]; acc.z += p[2]; acc.w += p[3];
  }
  const float inv = 1.0f / (float)lens[b];
  float* dst = pooled + ((size_t)b * LL + l) * DD + d0;
  dst[0] = acc.x * inv; dst[1] = acc.y * inv;
  dst[2] = acc.z * inv; dst[3] = acc.w * inv;
}

// ---------------------------------------------------------------------------
// Gate GEMM: hidden = tanh(pooled[208,1024] @ gate_w1[1024,256] + b1)
// fp32 WMMA 16x16x4, one wave (32 threads) per 16x16 output tile.
// grid: (13 M-tiles, 16 N-tiles), block: 32.
// A 16x4 frag: lanes 0-15 hold (K0,K1) of row m, lanes 16-31 hold (K2,K3).
// D 16x16 frag: element [v + 8*(lane>=16)][lane&15] in vgpr v.
// ---------------------------------------------------------------------------
__global__ __launch_bounds__(32) void gate_gemm_tanh_kernel(
    const float* __restrict__ pooled, const float* __restrict__ w1,
    const float* __restrict__ b1, float* __restrict__ hidden) {
  const int mt   = blockIdx.x;
  const int nt   = blockIdx.y;
  const int lane = threadIdx.x;
  const int hb   = lane >> 4;         // half-wave: 0 -> K0/K1, 1 -> K2/K3
  const int m    = lane & 15;

  const int row = mt * 16 + m;        // b-major row of pooled
  const int n   = nt * 16 + m;        // output column for B-frag / store

  v8f acc = {};
  for (int k0 = 0; k0 < DD; k0 += 4) {
    const int ka = k0 + hb * 2;
    v2f a, bv;
    a.x  = pooled[(size_t)row * DD + ka];
    a.y  = pooled[(size_t)row * DD + ka + 1];
    bv.x = w1[(size_t)ka * HH + n];
    bv.y = w1[(size_t)(ka + 1) * HH + n];
    acc = __builtin_amdgcn_wmma_f32_16x16x4_f32(
        /*neg_a=*/false, a, /*neg_b=*/false, bv,
        /*c_mod=*/(short)0, acc, /*reuse_a=*/false, /*reuse_b=*/false);
  }
#pragma unroll
  for (int v = 0; v < 8; ++v) {
    const int ro = mt * 16 + v + hb * 8;
    const int co = nt * 16 + m;
    hidden[(size_t)ro * HH + co] = tanhf(acc[v] + b1[co]);
  }
}

// ---------------------------------------------------------------------------
// Per-batch gate score reduce + softmax over L + weighted layer sum.
// grid: 16 (b), block: 256 (= HH).
// Writes final_emb -> d_out[OUT_FE ...], dyn weights -> d_out[OUT_DW ...].
// ---------------------------------------------------------------------------
__global__ __launch_bounds__(256) void gate_softmax_mix_kernel(
    const float* __restrict__ hidden, const float* __restrict__ w2,
    const float* __restrict__ pooled, float* __restrict__ out) {
  const int b = blockIdx.x;
  const int t = threadIdx.x;

  __shared__ float sred[HH];
  __shared__ float sc[LL];

  for (int l = 0; l < LL; ++l) {
    sred[t] = hidden[((size_t)b * LL + l) * HH + t] * w2[t];
    __syncthreads();
    for (int off = HH / 2; off > 0; off >>= 1) {
      if (t < off) sred[t] += sred[t + off];
      __syncthreads();
    }
    if (t == 0) sc[l] = sred[0];
    __syncthreads();
  }

  // redundant per-thread softmax over LL=13 (cheap, keeps everything in regs)
  float mx = sc[0];
  for (int l = 1; l < LL; ++l) mx = fmaxf(mx, sc[l]);
  float w[LL];
  float sum = 0.f;
  for (int l = 0; l < LL; ++l) { w[l] = __expf(sc[l] - mx); sum += w[l]; }
  const float inv = 1.0f / sum;
  for (int l = 0; l < LL; ++l) w[l] *= inv;

  if (t < LL) out[OUT_DW + b * LL + t] = w[t];

  for (int d = t; d < DD; d += HH) {
    float acc = 0.f;
    for (int l = 0; l < LL; ++l)
      acc += pooled[((size_t)b * LL + l) * DD + d] * w[l];
    out[OUT_FE + b * DD + d] = acc;
  }
}

// ---------------------------------------------------------------------------
// Projection GEMM: proj = relu(final_emb[16,1024] @ proj_w[1024,256] + b)
// Single M-tile (B=16 == M=16!). grid: 16 N-tiles, block: 32 (one wave).
// ---------------------------------------------------------------------------
__global__ __launch_bounds__(32) void proj_gemm_relu_kernel(
    const float* __restrict__ fe, const float* __restrict__ w,
    const float* __restrict__ bias, float* __restrict__ proj) {
  const int nt   = blockIdx.x;
  const int lane = threadIdx.x;
  const int hb   = lane >> 4;
  const int m    = lane & 15;
  const int n    = nt * 16 + m;

  v8f acc = {};
  for (int k0 = 0; k0 < DD; k0 += 4) {
    const int ka = k0 + hb * 2;
    v2f a, bv;
    a.x  = fe[(size_t)m * DD + ka];
    a.y  = fe[(size_t)m * DD + ka + 1];
    bv.x = w[(size_t)ka * HH + n];
    bv.y = w[(size_t)(ka + 1) * HH + n];
    acc = __builtin_amdgcn_wmma_f32_16x16x4_f32(
        false, a, false, bv, (short)0, acc, false, false);
  }
#pragma unroll
  for (int v = 0; v < 8; ++v) {
    const int ro = v + hb * 8;          // batch index
    const int co = nt * 16 + m;
    proj[(size_t)ro * HH + co] = fmaxf(acc[v] + bias[co], 0.0f);
  }
}

// ---------------------------------------------------------------------------
// Head: pred = proj[16,256] @ out_w[256,8] + out_b  -> d_out[0..127]
// grid: 1, block: 128 (= 16*8), one thread per output element.
// ---------------------------------------------------------------------------
__global__ __launch_bounds__(128) void head_kernel(
    const float* __restrict__ proj, const float* __restrict__ ow,
    const float* __restrict__ ob, float* __restrict__ out) {
  const int t = threadIdx.x;
  const int b = t >> 3;
  const int j = t & 7;
  float acc = ob[j];
  for (int h = 0; h < HH; ++h)
    acc += proj[(size_t)b * HH + h] * ow[(size_t)h * NLAB + j];
  out[OUT_PRED + b * NLAB + j] = acc;
}

// ---------------------------------------------------------------------------
extern "C" void kernel_launch(void* const* d_in, const int* in_sizes, int n_in,
                              void* d_out, int out_size, void* d_ws, size_t ws_size,
                              hipStream_t stream) {
  const float* feature = (const float*)d_in[0];
  const int*   lens    = (const int*)d_in[1];
  const float* gate_w1 = (const float*)d_in[2];
  const float* gate_b1 = (const float*)d_in[3];
  const float* gate_w2 = (const float*)d_in[4];
  const float* proj_w  = (const float*)d_in[5];
  const float* proj_b  = (const float*)d_in[6];
  const float* out_w   = (const float*)d_in[7];
  const float* out_b   = (const float*)d_in[8];
  float* out = (float*)d_out;

  // workspace layout (floats)
  float* ws       = (float*)d_ws;
  float* partials = ws;                                    // L*B*NCHUNK*D
  float* pooled   = partials + (size_t)LL * BB * NCHUNK * DD;  // L*B*D (b-major rows)
  float* hidden   = pooled + (size_t)LL * BB * DD;         // L*B*H
  float* proj     = hidden + (size_t)LL * BB * HH;         // B*H

  // 1) stream feature: partial masked sums (memory-bound, ~852 MB)
  pool_partial_kernel<<<dim3(LL * BB, NCHUNK), 256, 0, stream>>>(feature, lens, partials);
  // 2) reduce partials -> pooled (b-major)
  pool_reduce_kernel<<<LL * BB, 256, 0, stream>>>(partials, lens, pooled);
  // 3) gate MLP layer 1 (WMMA fp32) + tanh
  gate_gemm_tanh_kernel<<<dim3((LL * BB) / 16, HH / 16), 32, 0, stream>>>(
      pooled, gate_w1, gate_b1, hidden);
  // 4) gate score + softmax over layers + weighted sum -> final_emb, dyn_w
  gate_softmax_mix_kernel<<<BB, HH, 0, stream>>>(hidden, gate_w2, pooled, out);
  // 5) projection (WMMA fp32) + relu
  proj_gemm_relu_kernel<<<HH / 16, 32, 0, stream>>>(out + OUT_FE, proj_w, proj_b, proj);
  // 6) classification head -> pred
  head_kernel<<<1, 128, 0, stream>>>(proj, out_w, out_b, out);
}